// SigmoidContrastiveLearning_1374389534889
// MI455X (gfx1250) — compile-verified
//
#include <hip/hip_runtime.h>
#include <math.h>

typedef __attribute__((ext_vector_type(2))) float        v2f;
typedef __attribute__((ext_vector_type(8))) float        v8f;
typedef __attribute__((ext_vector_type(4))) unsigned int v4u;
typedef __attribute__((ext_vector_type(4))) int          v4i;
typedef __attribute__((ext_vector_type(8))) int          v8i;

#define N_TOK 8192
#define DIMD  128
#define BM    128
#define BN    128
// TDM pads 4 DWORDs every 128 DWORDs -> LDS row stride 132 floats.
// Bank for (row, col) = (132*row + col) % 64 = (4*row + col) % 64:
// rows 0..15 hit distinct banks, and the +2-column half-wave lands on the
// complementary set -> conflict-free ds_load_b64, 8B-aligned (528B rows).
#define LDS_STRIDE 132
#define LDS_TILE   (BM * LDS_STRIDE)          // floats per tile
#define LDS_BYTES  (size_t)(2 * LDS_TILE * 4) // 135168 B (< 320KB/WGP, 2 WGs/WGP)

// ---------------------------------------------------------------------------
// Issue one TDM descriptor: DMA a 128x128 fp32 tile (row stride DIMD) from
// global into LDS at lds_off, inserting 4-DWORD padding per 128-DWORD row.
// ---------------------------------------------------------------------------
__device__ __forceinline__ void tdm_load_tile(const float* gptr, unsigned lds_off)
{
    const unsigned long long ga = (unsigned long long)(uintptr_t)gptr;
    v4u g0;
    g0.x = 1u;                                   // count=1 (valid user D#)
    g0.y = lds_off;                              // lds_addr (bytes)
    g0.z = (unsigned)ga;                         // global_addr[31:0]
    g0.w = (unsigned)(ga >> 32) | (2u << 30);    // global_addr[56:32] | type=2
    v8i g1;
    g1[0] = 0x07920000;          // data_size=4B | pad_en | interval=128dw | pad=4dw
    g1[1] = (int)(DIMD  << 16);  // tensor_dim0 = 128  (bits 79:48, low half)
    g1[2] = (int)(N_TOK << 16);  // tensor_dim1 = 8192 (bits 111:80, low half)
    g1[3] = (int)(DIMD  << 16);  // tile_dim0   = 128  (bits 127:112)
    g1[4] = BM;                  // tile_dim1   = 128 rows (bits 143:128)
    g1[5] = DIMD;                // tensor_dim0_stride = 128 elems (bits 207:160)
    g1[6] = 0;
    g1[7] = 0;
    v4i z4 = {};
#if __clang_major__ >= 23
    v8i z8 = {};
    __builtin_amdgcn_tensor_load_to_lds(g0, g1, z4, z4, z8, 0);
#else
    __builtin_amdgcn_tensor_load_to_lds(g0, g1, z4, z4, 0);
#endif
}

// ---------------------------------------------------------------------------
// Fused GEMM (A @ B^T via V_WMMA_F32_16X16X4_F32, tiles TDM-staged in LDS)
// + SigLIP softplus epilogue. One 256-thread block = 8 wave32 per 128x128 tile.
// ---------------------------------------------------------------------------
__global__ __launch_bounds__(256) void siglip_tile_kernel(
    const float* __restrict__ A,      // audio_latents [N, D]
    const float* __restrict__ T,      // text_latents  [N, D]
    const float* __restrict__ tptr,   // log-temperature
    const float* __restrict__ bptr,   // bias
    float* __restrict__ partials)
{
    extern __shared__ float ldsbuf[];
    float* As = ldsbuf;               // [128][132]
    float* Ts = ldsbuf + LDS_TILE;    // [128][132]

    const int lane    = threadIdx.x & 31;
    const int wave    = threadIdx.x >> 5;   // 0..7
    const int waveRow = wave >> 2;          // 0..1 (64 rows each)
    const int waveCol = wave & 3;           // 0..3 (32 cols each)
    const int lane15  = lane & 15;
    const int laneHi  = lane >> 4;          // selects K pair {0,1} vs {2,3}

    const int rowBase = blockIdx.y * BM + waveRow * 64;
    const int colBase = blockIdx.x * BN + waveCol * 32;

    // ---- TDM stage: one wave issues both DMAs, everyone syncs --------------
    if (wave == 0) {
        tdm_load_tile(A + (size_t)blockIdx.y * BM * DIMD,
                      (unsigned)(uintptr_t)As);
        tdm_load_tile(T + (size_t)blockIdx.x * BN * DIMD,
                      (unsigned)(uintptr_t)Ts);
        __builtin_amdgcn_s_wait_tensorcnt(0);
    }
    __syncthreads();

    // ---- WMMA mainloop from LDS, double-buffered fragment prefetch --------
    v8f acc[4][2] = {};
    const float* Aw = As + (size_t)(waveRow * 64 + lane15) * LDS_STRIDE + 2 * laneHi;
    const float* Tw = Ts + (size_t)(waveCol * 32 + lane15) * LDS_STRIDE + 2 * laneHi;

    v2f afrag[2][4];
    v2f bfrag[2][2];
    #pragma unroll
    for (int m = 0; m < 4; ++m)
        afrag[0][m] = *(const v2f*)(Aw + m * 16 * LDS_STRIDE);
    #pragma unroll
    for (int n = 0; n < 2; ++n)
        bfrag[0][n] = *(const v2f*)(Tw + n * 16 * LDS_STRIDE);

    #pragma unroll
    for (int kk = 0; kk < DIMD / 4; ++kk) {
        const int cur = kk & 1;
        const int nxt = cur ^ 1;
        if (kk + 1 < DIMD / 4) {               // prefetch k+4 fragments
            const int kn = (kk + 1) * 4;
            #pragma unroll
            for (int m = 0; m < 4; ++m)
                afrag[nxt][m] = *(const v2f*)(Aw + m * 16 * LDS_STRIDE + kn);
            #pragma unroll
            for (int n = 0; n < 2; ++n)
                bfrag[nxt][n] = *(const v2f*)(Tw + n * 16 * LDS_STRIDE + kn);
        }
        #pragma unroll
        for (int m = 0; m < 4; ++m) {
            #pragma unroll
            for (int n = 0; n < 2; ++n) {
                acc[m][n] = __builtin_amdgcn_wmma_f32_16x16x4_f32(
                    false, afrag[cur][m], false, bfrag[cur][n],
                    (short)0, acc[m][n], false, false);
            }
        }
    }

    // ---- fused SigLIP epilogue (hardware exp/log softplus) ----------------
    const float scale = __expf(tptr[0]);
    const float biasv = bptr[0];

    float local = 0.0f;
    #pragma unroll
    for (int m = 0; m < 4; ++m) {
        #pragma unroll
        for (int n = 0; n < 2; ++n) {
            const int j = colBase + n * 16 + lane15;
            v8f c = acc[m][n];
            #pragma unroll
            for (int r = 0; r < 8; ++r) {
                const int i = rowBase + m * 16 + r + 8 * laneHi;
                float z = fmaf(c[r], scale, biasv);
                float x = (i == j) ? -z : z;           // label=+1 diag, -1 off
                // softplus(x) = max(x,0) + log(1 + exp(-|x|)); t in (0,1] so
                // plain __logf(1+t) is safe and maps to v_exp/v_log hardware.
                local += fmaxf(x, 0.0f) + __logf(1.0f + __expf(-fabsf(x)));
            }
        }
    }

    // ---- wave32 shuffle reduce, then cross-wave via LDS -------------------
    #pragma unroll
    for (int off = 16; off > 0; off >>= 1)
        local += __shfl_down(local, off, 32);

    __shared__ float red[8];
    if (lane == 0) red[wave] = local;
    __syncthreads();

    if (threadIdx.x == 0) {
        float s = 0.0f;
        #pragma unroll
        for (int w = 0; w < 8; ++w) s += red[w];
        partials[blockIdx.y * gridDim.x + blockIdx.x] = s;
    }
}

// ---------------------------------------------------------------------------
// Deterministic final reduction of block partials -> scalar loss.
// ---------------------------------------------------------------------------
__global__ __launch_bounds__(256) void siglip_reduce_kernel(
    const float* __restrict__ partials, float* __restrict__ out,
    int nparts, float inv_n)
{
    __shared__ float sm[256];
    float s = 0.0f;
    for (int i = threadIdx.x; i < nparts; i += 256) s += partials[i];
    sm[threadIdx.x] = s;
    __syncthreads();
    #pragma unroll
    for (int st = 128; st > 0; st >>= 1) {
        if ((int)threadIdx.x < st) sm[threadIdx.x] += sm[threadIdx.x + st];
        __syncthreads();
    }
    if (threadIdx.x == 0) out[0] = sm[0] * inv_n;
}

extern "C" void kernel_launch(void* const* d_in, const int* in_sizes, int n_in,
                              void* d_out, int out_size, void* d_ws, size_t ws_size,
                              hipStream_t stream) {
    const float* audio = (const float*)d_in[0];
    const float* text  = (const float*)d_in[1];
    const float* temp  = (const float*)d_in[2];
    const float* bias  = (const float*)d_in[3];
    float* out = (float*)d_out;
    float* ws  = (float*)d_ws;

    dim3 grid(N_TOK / BN, N_TOK / BM);   // 64 x 64 = 4096 tiles
    siglip_tile_kernel<<<grid, 256, LDS_BYTES, stream>>>(audio, text, temp, bias, ws);
    siglip_reduce_kernel<<<1, 256, 0, stream>>>(ws, out, (int)(grid.x * grid.y),
                                                1.0f / (float)N_TOK);
}